// R_22445499089387
// MI455X (gfx1250) — compile-verified
//
#include <hip/hip_runtime.h>

// ---------------------------------------------------------------------------
// ReLU-RNN (T=64, N=8192, F=H=256) + LayerNorm, fp32 via V_WMMA_F32_16X16X4_F32
// Kernel 1: Xw[t,n,:] = X[t,n,:] @ W_ih^T + b_ih        (staged into d_out)
// Kernel 2: persistent per-32-rows recurrence:
//           h = relu(Xw[t] + h @ W_hh^T + b_hh), LayerNorm(h) -> d_out[t]
// W matrices + h state held in LDS (CDNA5: 320KB/WGP), padded to kill bank
// conflicts on the WMMA fragment gathers.  Bulk staging uses the CDNA5 async
// memory->LDS path (GLOBAL_LOAD_ASYNC_TO_LDS_B128, ASYNCcnt).
// ---------------------------------------------------------------------------

typedef float v2f __attribute__((ext_vector_type(2)));
typedef float v4f __attribute__((ext_vector_type(4)));
typedef float v8f __attribute__((ext_vector_type(8)));

#define TT   64
#define NB   8192
#define FD   256
#define HD   256
#define LDSP 260          // padded LDS row stride in dwords (260 % 64 == 4 -> conflict-free b64 gathers)
#define BM   32           // batch rows per block tile
#define LN_EPS 1e-5f

// ---- CDNA5 async memory->LDS copy (16B per lane), tracked by ASYNCcnt -------
__device__ __forceinline__ void async_copy_b128(const void* gbase, unsigned goff_bytes,
                                                unsigned lds_off_bytes) {
  // GVS mode: mem_addr = SADDR + VGPR_offset + inst_offset; LDS dest from VDST vgpr
  asm volatile("global_load_async_to_lds_b128 %0, %1, %2"
               :
               : "v"(lds_off_bytes), "v"(goff_bytes), "s"(gbase)
               : "memory");
}

__device__ __forceinline__ void async_wait0() {
#if __has_builtin(__builtin_amdgcn_s_wait_asynccnt)
  __builtin_amdgcn_s_wait_asynccnt(0);
#else
  asm volatile("s_wait_asynccnt 0x0" ::: "memory");
#endif
}

// LDS byte offset of a __shared__ element (flat LDS addr low 32 bits == offset)
__device__ __forceinline__ unsigned lds_off(const void* p) {
  return (unsigned)(unsigned long long)p;
}

// ---- one 16x64 strip of C += A(16xK) * B(K x 64), A rows from sA, B=W^T from sW
// wave layout: wrow in {0,1} selects 16-row group, wcol in {0..3} selects 64-col group
__device__ __forceinline__ void gemm_strip(const float* __restrict__ sA,
                                           const float* __restrict__ sW,
                                           int wrow, int wcol, int lm, int lhi,
                                           v8f acc[4]) {
  const int arow = wrow * 16 + lm;            // A row for this lane
  const float* aRow = sA + arow * LDSP;
#pragma unroll 8
  for (int k0 = 0; k0 < HD; k0 += 4) {
    // A 16x4 f32 fragment: lane L holds A[M=L%16][k0+2*(L/16) .. +1] in 2 VGPRs
    v2f a = *(const v2f*)(aRow + k0 + 2 * lhi);
#pragma unroll
    for (int j = 0; j < 4; ++j) {
      // B 4x16 fragment, B[k][n] = W[n][k]: lane L holds W[n0+L%16][k0+2*(L/16) .. +1]
      const int n = wcol * 64 + j * 16 + lm;
      v2f b = *(const v2f*)(sW + n * LDSP + k0 + 2 * lhi);
      acc[j] = __builtin_amdgcn_wmma_f32_16x16x4_f32(
          /*neg_a=*/false, a, /*neg_b=*/false, b,
          /*c_mod=*/(short)0, acc[j], /*reuse_a=*/false, /*reuse_b=*/false);
    }
  }
}

// ---------------------------------------------------------------------------
// Kernel 1: input projection.  Xw (staged in d_out) = X @ W_ih^T + b_ih
// ---------------------------------------------------------------------------
__global__ __launch_bounds__(256) void k_inproj(const float* __restrict__ X,
                                                const float* __restrict__ Wih,
                                                const float* __restrict__ bih,
                                                float* __restrict__ Xw,
                                                int num_tiles) {
  __shared__ float sW[HD * LDSP];   // 260 KB
  __shared__ float sX[BM * LDSP];   // 32.5 KB

  const int tid = threadIdx.x;

  // async-stage W_ih [256][256] -> padded LDS (16B per lane per op)
#pragma unroll
  for (int i = 0; i < (HD * FD) / (256 * 4); ++i) {
    const int e = (i * 256 + tid) * 4;
    const int r = e >> 8, c = e & 255;
    async_copy_b128(Wih, (unsigned)e * 4u, lds_off(&sW[r * LDSP + c]));
  }

  const int lane = tid & 31;
  const int wave = tid >> 5;
  const int wrow = wave >> 2;   // 0..1
  const int wcol = wave & 3;    // 0..3
  const int lm   = lane & 15;
  const int lhi  = lane >> 4;

  // per-lane output-column bias (fixed across row tiles)
  float biasj[4];
#pragma unroll
  for (int j = 0; j < 4; ++j) biasj[j] = bih[wcol * 64 + j * 16 + lm];

  for (int tile = blockIdx.x; tile < num_tiles; tile += gridDim.x) {
    const long long row0 = (long long)tile * BM;

    // async-stage X tile: rows are contiguous in memory (32*256 floats = 32KB)
    const float* src = X + row0 * FD;
#pragma unroll
    for (int i = 0; i < (BM * FD) / (256 * 4); ++i) {
      const int e = (i * 256 + tid) * 4;
      const int r = e >> 8, c = e & 255;
      async_copy_b128(src, (unsigned)e * 4u, lds_off(&sX[r * LDSP + c]));
    }
    async_wait0();      // covers W (first iteration) + this X tile
    __syncthreads();

    v8f acc[4];
#pragma unroll
    for (int j = 0; j < 4; ++j)
#pragma unroll
      for (int v = 0; v < 8; ++v) acc[j][v] = biasj[j];

    gemm_strip(sX, sW, wrow, wcol, lm, lhi, acc);

    // store C fragments: lane L, vgpr v -> row M = v + 8*(L/16), col N = L%16
#pragma unroll
    for (int j = 0; j < 4; ++j) {
      const int col = wcol * 64 + j * 16 + lm;
#pragma unroll
      for (int v = 0; v < 8; ++v) {
        const long long r = row0 + wrow * 16 + v + 8 * lhi;
        Xw[r * HD + col] = acc[j][v];
      }
    }
    __syncthreads();   // protect sX before next tile's staging
  }
}

// ---------------------------------------------------------------------------
// Kernel 2: recurrence + LayerNorm.  Each block owns BM=32 batch rows for all T.
// ---------------------------------------------------------------------------
__global__ __launch_bounds__(256) void k_recurrent(const float* __restrict__ Whh,
                                                   const float* __restrict__ bhh,
                                                   const float* __restrict__ lnw,
                                                   const float* __restrict__ lnb,
                                                   float* __restrict__ Y) {
  __shared__ float sW[HD * LDSP];   // W_hh, 260 KB
  __shared__ float sH[BM * LDSP];   // h state, 32.5 KB

  const int tid = threadIdx.x;
  const long long row0 = (long long)blockIdx.x * BM;

  // async-stage W_hh -> LDS
#pragma unroll
  for (int i = 0; i < (HD * HD) / (256 * 4); ++i) {
    const int e = (i * 256 + tid) * 4;
    const int r = e >> 8, c = e & 255;
    async_copy_b128(Whh, (unsigned)e * 4u, lds_off(&sW[r * LDSP + c]));
  }

  // async-stage Xw[0] tile (contiguous in Y) -> sH, then add b_hh in place
  const float* xw0 = Y + row0 * HD;
#pragma unroll
  for (int i = 0; i < (BM * HD) / (256 * 4); ++i) {
    const int e = (i * 256 + tid) * 4;
    const int r = e >> 8, c = e & 255;
    async_copy_b128(xw0, (unsigned)e * 4u, lds_off(&sH[r * LDSP + c]));
  }
  async_wait0();
  __syncthreads();

  // h0 = Xw[0] + b_hh
#pragma unroll
  for (int i = 0; i < (BM * HD) / (256 * 4); ++i) {
    const int e = (i * 256 + tid) * 4;
    const int r = e >> 8, c = e & 255;
    v4f x = *(const v4f*)(&sH[r * LDSP + c]);
    v4f b = *(const v4f*)(bhh + c);
    *(v4f*)(&sH[r * LDSP + c]) = x + b;
  }
  __syncthreads();

  const int lane = tid & 31;
  const int wave = tid >> 5;
  const int wrow = wave >> 2;
  const int wcol = wave & 3;
  const int lm   = lane & 15;
  const int lhi  = lane >> 4;

  float biasj[4];
#pragma unroll
  for (int j = 0; j < 4; ++j) biasj[j] = bhh[wcol * 64 + j * 16 + lm];

  // LN assignment: wave handles 4 rows; 8 lanes per row, 32 cols per lane
  const int ln_row = wave * 4 + (lane >> 3);
  const int ln_cg  = lane & 7;

  for (int t = 0; t < TT; ++t) {
    const float* xwt = Y + ((long long)t * NB + row0) * HD;

    // prefetch next step's Xw tile (32KB contiguous) while we compute
    if (t + 1 < TT)
      __builtin_prefetch(Y + ((long long)(t + 1) * NB + row0) * HD + tid * 32, 0, 0);

    // acc = Xw[t] fragment + b_hh (C-layout gather: 64B-coalesced per vgpr)
    v8f acc[4];
#pragma unroll
    for (int j = 0; j < 4; ++j) {
      const int col = wcol * 64 + j * 16 + lm;
#pragma unroll
      for (int v = 0; v < 8; ++v)
        acc[j][v] = xwt[(long long)(wrow * 16 + v + 8 * lhi) * HD + col] + biasj[j];
    }

    // acc += h @ W_hh^T
    gemm_strip(sH, sW, wrow, wcol, lm, lhi, acc);

    // relu
#pragma unroll
    for (int j = 0; j < 4; ++j)
#pragma unroll
      for (int v = 0; v < 8; ++v) acc[j][v] = fmaxf(acc[j][v], 0.0f);

    __syncthreads();   // all waves done reading old h (and Xw reads retired via acc deps)

    // write h_new into LDS (conflict-free: 16 consecutive cols per vgpr)
#pragma unroll
    for (int j = 0; j < 4; ++j) {
      const int col = wcol * 64 + j * 16 + lm;
#pragma unroll
      for (int v = 0; v < 8; ++v)
        sH[(wrow * 16 + v + 8 * lhi) * LDSP + col] = acc[j][v];
    }
    __syncthreads();

    // -------- LayerNorm over H=256 for this block's 32 rows --------
    float s1 = 0.0f, s2 = 0.0f;
    const float* hrow = sH + ln_row * LDSP;
#pragma unroll
    for (int c = ln_cg * 32; c < ln_cg * 32 + 32; ++c) {
      const float h = hrow[c];
      s1 += h;
      s2 += h * h;
    }
    // reduce across the 8 lanes covering this row (wave32 shuffles)
#pragma unroll
    for (int m = 1; m < 8; m <<= 1) {
      s1 += __shfl_xor(s1, m, 32);
      s2 += __shfl_xor(s2, m, 32);
    }
    const float mean = s1 * (1.0f / HD);
    const float var  = s2 * (1.0f / HD) - mean * mean;
    const float rstd = rsqrtf(var + LN_EPS);

    float* outr = Y + ((long long)t * NB + row0 + ln_row) * HD;
#pragma unroll
    for (int c4 = ln_cg * 32; c4 < ln_cg * 32 + 32; c4 += 4) {
      v4f h = *(const v4f*)(hrow + c4);
      v4f w = *(const v4f*)(lnw + c4);
      v4f b = *(const v4f*)(lnb + c4);
      v4f o;
#pragma unroll
      for (int e = 0; e < 4; ++e) o[e] = (h[e] - mean) * rstd * w[e] + b[e];
      *(v4f*)(outr + c4) = o;
    }
    // next iteration's pre-write __syncthreads protects sH against these reads
  }
}

// ---------------------------------------------------------------------------
extern "C" void kernel_launch(void* const* d_in, const int* in_sizes, int n_in,
                              void* d_out, int out_size, void* d_ws, size_t ws_size,
                              hipStream_t stream) {
  (void)in_sizes; (void)n_in; (void)d_ws; (void)ws_size; (void)out_size;
  const float* X   = (const float*)d_in[0];
  const float* Wih = (const float*)d_in[1];
  const float* Whh = (const float*)d_in[2];
  const float* bih = (const float*)d_in[3];
  const float* bhh = (const float*)d_in[4];
  const float* lnw = (const float*)d_in[5];
  const float* lnb = (const float*)d_in[6];
  float* out = (float*)d_out;

  const int num_tiles = (TT * NB) / BM;          // 16384 row tiles
  k_inproj<<<2048, 256, 0, stream>>>(X, Wih, bih, out, num_tiles);
  k_recurrent<<<NB / BM, 256, 0, stream>>>(Whh, bhh, lnw, lnb, out);
}